// Depth_to_pts_72825465471374
// MI455X (gfx1250) — compile-verified
//
#include <hip/hip_runtime.h>
#include <stdint.h>

// Problem constants (fixed by the reference's setup_inputs()).
#define BS        16
#define H         384
#define W         1248
#define NPIX      (H * W)          // 479232 = 1872 * 256 exactly
#define NPOINTS   16384
#define FAR_NUM   3277
#define NEAR_TH   40.0f
#define SCALE_S   0.8f
#define NBLK      (NPIX / 256)     // 1872 per batch
#define TILES     (W / 16)         // 78 WMMA tiles per row

typedef float v2f __attribute__((ext_vector_type(2)));
typedef float v8f __attribute__((ext_vector_type(8)));

// ---------------------------------------------------------------------------
// Kernel 1: zoom-gather + WMMA back-projection.
// One block (8 wave32) per (batch,row). Each wave processes 16 pixels per
// V_WMMA_F32_16X16X4_F32: A(16x4) = [x*d, y*d, d, 1] per point, B(4x16) holds
// the per-batch projection in columns 0..2, D(16x16) f32 -> columns 0..2 are
// (x,y,z). Lanes 0-15 carry K=0,1 ; lanes 16-31 carry K=2,3 (ISA A layout).
// D layout: lane holds N = lane&15, rows M = v + 8*(lane>>4) across 8 VGPRs.
//
// The nearest-neighbor resample index folds to a single FMA:
//   ix = rint(s*px + Cx),  Cx = (s + (tx + 1 - s)*W - 1) * 0.5
// (same for iy), so the inner loop has no division.
// ---------------------------------------------------------------------------
__global__ __launch_bounds__(256) void proj_kernel(
    const float* __restrict__ depth, const float* __restrict__ P2,
    const int* __restrict__ pDBHx, const int* __restrict__ pDBHy,
    const int* __restrict__ pSx,   const int* __restrict__ pSy,
    float* __restrict__ pts)
{
    const int bid  = blockIdx.x;
    const int b    = bid / H;
    const int row  = bid - b * H;
    const int tid  = threadIdx.x;
    const int wave = tid >> 5;
    const int lane = tid & 31;
    const int ncol = lane & 15;   // N column of B/D this lane owns
    const int hi   = lane >> 4;   // 0: K=0,1  1: K=2,3

    const float dbx = (float)pDBHx[0];
    const float dby = (float)pDBHy[0];
    const float stx = (float)pSx[0];
    const float sty = (float)pSy[0];

    const float s    = SCALE_S;
    const float zoom = 1.0f - s;
    const float tx   = zoom * dbx / (float)W - zoom * 0.5f;
    const float ty   = zoom * dby / (float)H - zoom * 0.5f;
    const float xoff = (zoom / s) * dbx + stx;
    const float yoff = (zoom / s) * dby + sty;

    // Folded resample constants: ix = rint(s*px + Cx), iy = rint(s*row + Cy)
    const float Cx = (s + (tx + 1.0f - s) * (float)W - 1.0f) * 0.5f;
    const float Cy = (s + (ty + 1.0f - s) * (float)H - 1.0f) * 0.5f;

    // Per-batch camera parameters (zp = P2 * diag(1/s,1/s,1) row-scaling).
    const float* p   = P2 + (size_t)b * 12;
    const float invs = 1.0f / s;
    const float fu = p[0] * invs, cu = p[2] * invs;
    const float fv = p[5] * invs, cv = p[6] * invs;
    const float bx = -p[3] / p[0];          // scale cancels in p[3]/p[0]
    const float by = -p[7] / p[5];

    // B operand: column ncol of the 4x3 projection (cols >=3 are zero).
    // col0 (x): [1/fu, 0, (xoff-cu)/fu, bx]
    // col1 (y): [0, 1/fv, (yoff-cv)/fv, by]
    // col2 (z): [0, 0, 1, 0]
    const float w0 = (ncol == 0) ? (1.0f / fu) : 0.0f;
    const float w1 = (ncol == 1) ? (1.0f / fv) : 0.0f;
    const float w2 = (ncol == 0) ? ((xoff - cu) / fu)
                   : (ncol == 1) ? ((yoff - cv) / fv)
                   : (ncol == 2) ? 1.0f : 0.0f;
    const float w3 = (ncol == 0) ? bx : (ncol == 1) ? by : 0.0f;
    v2f Bm;
    Bm.x = hi ? w2 : w0;   // K=2 : K=0
    Bm.y = hi ? w3 : w1;   // K=3 : K=1

    // Source row for this output row (round-to-even matches jnp.round).
    int iy = (int)__builtin_rintf(__builtin_fmaf(s, (float)row, Cy));
    iy = min(max(iy, 0), H - 1);
    const float* __restrict__ srow = depth + ((size_t)b * H + (size_t)iy) * W;
    float* __restrict__ orow = pts + ((size_t)b * NPIX + (size_t)row * W) * 3;

    const float yf = (float)row;
    const int m = lane & 15;

    for (int it = 0; it < (TILES + 7) / 8; ++it) {
        int tile = wave + it * 8;
        const bool valid = (tile < TILES);
        tile = valid ? tile : 0;          // select, keeps EXEC all-ones

        const int px = tile * 16 + m;
        int ix = (int)__builtin_rintf(__builtin_fmaf(s, (float)px, Cx));
        ix = min(max(ix, 0), W - 1);
        const float d = srow[ix];                       // gather (cache-local)

        // Prefetch ~8 tiles ahead in the source row (0.8 * 128 cols).
        __builtin_prefetch(srow + min(ix + 102, W - 1), 0, 1);

        v2f Am;
        Am.x = hi ? d    : (float)px * d;   // K=2 : K=0
        Am.y = hi ? 1.0f : yf * d;          // K=3 : K=1

        v8f cz = {};
        v8f Dm = __builtin_amdgcn_wmma_f32_16x16x4_f32(
            /*neg_a=*/false, Am, /*neg_b=*/false, Bm,
            /*c_mod=*/(short)0, cz, /*reuse_a=*/false, /*reuse_b=*/false);

        // Lanes owning columns 0..2 write (x,y,z) interleaved per point.
        if (valid && ncol < 3) {
            float* basep = orow + (size_t)(tile * 16 + hi * 8) * 3 + ncol;
            #pragma unroll
            for (int v = 0; v < 8; ++v)
                basep[v * 3] = Dm[v];
        }
    }
}

// ---------------------------------------------------------------------------
// Kernel 2: per-256-element near counts (z < 40). LDS tree reduction.
// ---------------------------------------------------------------------------
__global__ __launch_bounds__(256) void count_kernel(
    const float* __restrict__ pts, int* __restrict__ blockCounts)
{
    __shared__ int sm[256];
    const int b   = blockIdx.x / NBLK;
    const int blk = blockIdx.x - b * NBLK;
    const int li  = threadIdx.x;
    const int i   = blk * 256 + li;
    const float z = pts[((size_t)b * NPIX + i) * 3 + 2];
    sm[li] = (z < NEAR_TH) ? 1 : 0;
    __syncthreads();
    for (int off = 128; off > 0; off >>= 1) {
        if (li < off) sm[li] += sm[li + off];
        __syncthreads();
    }
    if (li == 0) blockCounts[b * NBLK + blk] = sm[0];
}

// ---------------------------------------------------------------------------
// Kernel 3: per-batch exclusive scan of block counts + take computation.
// meta[b] = {near_count, near_take, far_take, far_count}
// ---------------------------------------------------------------------------
__global__ void scan_kernel(const int* __restrict__ blockCounts,
                            int* __restrict__ blockOffsets,
                            int* __restrict__ meta)
{
    const int b = blockIdx.x;
    if (threadIdx.x == 0) {
        int acc = 0;
        for (int i = 0; i < NBLK; ++i) {
            blockOffsets[b * NBLK + i] = acc;
            acc += blockCounts[b * NBLK + i];
        }
        const int nearc = acc;
        const int farc  = NPIX - nearc;
        const int fart  = min(farc, FAR_NUM);
        const int neart = NPOINTS - fart;
        meta[b * 4 + 0] = nearc;
        meta[b * 4 + 1] = neart;
        meta[b * 4 + 2] = fart;
        meta[b * 4 + 3] = farc;
    }
}

// ---------------------------------------------------------------------------
// Kernel 4: stream compaction. Near indices packed ascending from the front
// of list[b], far indices packed from the back — one buffer, fully
// deterministic (LDS Hillis-Steele scan; no ordering atomics).
// ---------------------------------------------------------------------------
__global__ __launch_bounds__(256) void compact_kernel(
    const float* __restrict__ pts, const int* __restrict__ blockOffsets,
    int* __restrict__ list)
{
    __shared__ int sc[256];
    const int b   = blockIdx.x / NBLK;
    const int blk = blockIdx.x - b * NBLK;
    const int li  = threadIdx.x;
    const int i   = blk * 256 + li;
    const float z = pts[((size_t)b * NPIX + i) * 3 + 2];
    const int flag = (z < NEAR_TH) ? 1 : 0;
    sc[li] = flag;
    __syncthreads();
    for (int off = 1; off < 256; off <<= 1) {
        const int v   = sc[li];
        const int add = (li >= off) ? sc[li - off] : 0;
        __syncthreads();
        sc[li] = v + add;
        __syncthreads();
    }
    const int excl    = sc[li] - flag;             // exclusive near rank
    const int nearOff = blockOffsets[b * NBLK + blk];
    int* __restrict__ L = list + (size_t)b * NPIX;
    if (flag) {
        L[nearOff + excl] = i;
    } else {
        const int farOff = blk * 256 - nearOff;    // far elements before block
        const int farIdx = farOff + (li - excl);
        L[NPIX - 1 - farIdx] = i;
    }
}

// ---------------------------------------------------------------------------
// Kernel 5: sample NPOINTS per batch. Bijective LCG permutation on 14 bits
// shuffles near/far slots; distinct strided pick from each compacted list.
// ---------------------------------------------------------------------------
__global__ __launch_bounds__(256) void sample_kernel(
    const float* __restrict__ pts, const int* __restrict__ list,
    const int* __restrict__ meta, float* __restrict__ out)
{
    const int b = blockIdx.x / (NPOINTS / 256);
    const int p = (blockIdx.x % (NPOINTS / 256)) * 256 + threadIdx.x;

    const int nearc = meta[b * 4 + 0];
    const int neart = meta[b * 4 + 1];
    const int fart  = meta[b * 4 + 2];
    const int farc  = meta[b * 4 + 3];

    // Bijection on [0, 16384): odd multiplier LCG + batch-hashed offset.
    const unsigned q = (24317u * (unsigned)p +
                        (((unsigned)b * 2654435761u) >> 18)) & (NPOINTS - 1);

    const int* __restrict__ L = list + (size_t)b * NPIX;
    int src;
    if ((int)q < neart) {
        long long j = ((long long)q * (long long)max(nearc, 1)) /
                      (long long)max(neart, 1);
        j = min((long long)max(nearc - 1, 0), j);
        src = L[(int)j];
    } else {
        const int fq = (int)q - neart;
        long long j = ((long long)fq * (long long)max(farc, 1)) /
                      (long long)max(fart, 1);
        j = min((long long)max(farc - 1, 0), j);
        src = L[NPIX - 1 - (int)j];
    }

    const float* sp = pts + ((size_t)b * NPIX + (size_t)src) * 3;
    float* op = out + ((size_t)b * NPOINTS + (size_t)p) * 3;
    op[0] = sp[0];
    op[1] = sp[1];
    op[2] = sp[2];
}

// ---------------------------------------------------------------------------
extern "C" void kernel_launch(void* const* d_in, const int* in_sizes, int n_in,
                              void* d_out, int out_size, void* d_ws,
                              size_t ws_size, hipStream_t stream)
{
    (void)in_sizes; (void)n_in; (void)out_size; (void)ws_size;

    const float* depth = (const float*)d_in[0];
    const float* P2    = (const float*)d_in[1];
    const int*   DBHx  = (const int*)d_in[2];
    const int*   DBHy  = (const int*)d_in[3];
    const int*   sx    = (const int*)d_in[4];
    const int*   sy    = (const int*)d_in[5];
    float*       out   = (float*)d_out;

    char* ws = (char*)d_ws;
    size_t off = 0;
    float* pts = (float*)(ws + off);         off += (size_t)BS * NPIX * 3 * sizeof(float);
    int* list  = (int*)(ws + off);           off += (size_t)BS * NPIX * sizeof(int);
    int* blockCounts  = (int*)(ws + off);    off += (size_t)BS * NBLK * sizeof(int);
    int* blockOffsets = (int*)(ws + off);    off += (size_t)BS * NBLK * sizeof(int);
    int* meta = (int*)(ws + off);

    proj_kernel<<<BS * H, 256, 0, stream>>>(depth, P2, DBHx, DBHy, sx, sy, pts);
    count_kernel<<<BS * NBLK, 256, 0, stream>>>(pts, blockCounts);
    scan_kernel<<<BS, 32, 0, stream>>>(blockCounts, blockOffsets, meta);
    compact_kernel<<<BS * NBLK, 256, 0, stream>>>(pts, blockOffsets, list);
    sample_kernel<<<BS * (NPOINTS / 256), 256, 0, stream>>>(pts, list, meta, out);
}